// DifferentiableSimulator_80874234184044
// MI455X (gfx1250) — compile-verified
//
#include <hip/hip_runtime.h>
#include <hip/hip_bf16.h>
#include <math.h>

// CDNA5 wave32 WMMA types: f32 16x16x4 -> A,B = 2 VGPRs (v2f), C/D = 8 VGPRs (v8f)
typedef __attribute__((ext_vector_type(2))) float v2f;
typedef __attribute__((ext_vector_type(8))) float v8f;

#define NW     16                 // workgroups (one per WGP), partition hidden dim
#define TSTEPS 50000
#define SDIM   256
#define UDIM   64
#define HDIM   1024
#define YDIM   128
#define ZDIM   (SDIM + UDIM)      // 320
#define HS     (HDIM / NW)        // 64 hidden units per WGP
#define PDIM   (SDIM + YDIM)      // 384 combined [rhs | y] partial
#define DT_STEP 0.01f

__global__ void init_sync(unsigned* ctr) {
  if (threadIdx.x < 2) ctr[threadIdx.x] = 0u;
}

__launch_bounds__(256)
__global__ void ode_scan(const float* __restrict__ x_init,
                         const float* __restrict__ u,
                         const float* __restrict__ W1,    // [320][1024]
                         const float* __restrict__ b1,    // [1024]
                         const float* __restrict__ W2r,   // [1024][256]
                         const float* __restrict__ b2r,   // [256]
                         const float* __restrict__ W2y,   // [1024][128]
                         const float* __restrict__ b2y,   // [128]
                         float* __restrict__ out,         // xsol[T][256] ++ ysol[T][128]
                         float* __restrict__ gpart,       // [2][NW][384] partial exchange
                         unsigned* __restrict__ ctr)      // [2] barrier counters
{
  __shared__ __align__(32) float zs[ZDIM];      // current [x | u_i]
  __shared__ __align__(32) float hpart[2 * HS]; // k-slice partials of hidden pre-act
  __shared__ __align__(32) float hs[HS];        // tanh(hidden) slice
  __shared__ __align__(32) float pout[PDIM];    // this WGP's [rhs|y] partial
  __shared__ float b1s[HS];
  __shared__ float b2rs[SDIM];
  __shared__ float b2ys[YDIM];

  const int tid  = threadIdx.x;
  const int g    = blockIdx.x;      // which hidden slice we own
  const int lane = tid & 31;
  const int wave = tid >> 5;        // 8 waves

  float* xout = out;
  float* yout = out + (size_t)TSTEPS * SDIM;

  // ---- biases + initial state into LDS ----
  if (tid < SDIM) { zs[tid] = x_init[tid]; b2rs[tid] = b2r[tid]; }
  if (tid < HS)   b1s[tid]  = b1[g * HS + tid];
  if (tid < YDIM) b2ys[tid] = b2y[tid];

  // ---- weight tiles -> registers, in WMMA **A** layout (M x K = 16 x 4):
  //   lanes 0-15 : VGPR0=A[m][0], VGPR1=A[m][1] (m=lane); lanes 16-31 : A[m][2], A[m][3]
  // with M = output column of the GEMV, K = contraction index.
  const int kk = 2 * (lane >> 4);   // 0 or 2
  const int nn = lane & 15;         // M within tile
  const int hi = (lane >> 4) << 3;  // D rows 0-7 live in lane group 0, rows 8-15 in group 1

  // GEMV1 (z[320] -> h slice[64]): wave -> (column chunk c1 of 4, k-slice s1 of 2x160)
  const int c1 = wave & 3;
  const int s1 = wave >> 2;
  const int col1 = g * HS + c1 * 16 + nn;

  v2f w1t[40];
#pragma unroll
  for (int t = 0; t < 40; ++t) {
    int k0 = s1 * 160 + t * 4 + kk;
    w1t[t].x = W1[(size_t)k0 * HDIM + col1];
    w1t[t].y = W1[(size_t)(k0 + 1) * HDIM + col1];
  }

  // GEMV2 (h slice[64] -> combined [rhs(256)|y(128)]): wave handles 3 of 24 column chunks
  v2f w2t[48];
#pragma unroll
  for (int cc = 0; cc < 3; ++cc) {
    int n0g = (wave * 3 + cc) * 16;
#pragma unroll
    for (int t = 0; t < 16; ++t) {
      int k0   = t * 4 + kk;
      int row0 = g * HS + k0;
      float a0, a1;
      if (n0g < SDIM) {
        a0 = W2r[(size_t)row0 * SDIM + n0g + nn];
        a1 = W2r[(size_t)(row0 + 1) * SDIM + n0g + nn];
      } else {
        a0 = W2y[(size_t)row0 * YDIM + (n0g - SDIM) + nn];
        a1 = W2y[(size_t)(row0 + 1) * YDIM + (n0g - SDIM) + nn];
      }
      w2t[cc * 16 + t].x = a0;
      w2t[cc * 16 + t].y = a1;
    }
  }
  __syncthreads();

#pragma unroll 1
  for (int i = 0; i < TSTEPS; ++i) {
    // stream u_i, record pre-update x, prefetch ahead
    if (tid < UDIM) zs[SDIM + tid] = u[(size_t)i * UDIM + tid];
    if (g == 0)     xout[(size_t)i * SDIM + tid] = zs[tid];
    if (tid == 0 && i + 8 < TSTEPS) __builtin_prefetch(&u[(size_t)(i + 8) * UDIM], 0, 0);
    __syncthreads();

    // ---- GEMV1: A = resident W1 tile, B = z chunk broadcast into all 16 columns.
    //      One ds_load_b64 + one v_wmma per K-chunk; no lane masking needed.
    v8f acc = {};
#pragma unroll
    for (int t = 0; t < 40; ++t) {
      int k0 = s1 * 160 + t * 4;
      v2f bz = *(const v2f*)(zs + k0 + kk);   // lanes 0-15: K0,K1; lanes 16-31: K2,K3
      acc = __builtin_amdgcn_wmma_f32_16x16x4_f32(false, w1t[t], false, bz,
                                                  (short)0, acc, false, false);
    }
    // every column of D holds the GEMV result; lane 0 carries rows 0-7, lane 16 rows 8-15
    if (nn == 0) *(v8f*)(hpart + s1 * HS + c1 * 16 + hi) = acc;
    __syncthreads();

    if (tid < HS) hs[tid] = tanhf(hpart[tid] + hpart[HS + tid] + b1s[tid]);
    __syncthreads();

    // ---- GEMV2: same trick, contraction over the 64 local hidden units ----
#pragma unroll
    for (int cc = 0; cc < 3; ++cc) {
      int n0g = (wave * 3 + cc) * 16;
      v8f acc2 = {};
#pragma unroll
      for (int t = 0; t < 16; ++t) {
        v2f bh = *(const v2f*)(hs + t * 4 + kk);
        acc2 = __builtin_amdgcn_wmma_f32_16x16x4_f32(false, w2t[cc * 16 + t], false, bh,
                                                     (short)0, acc2, false, false);
      }
      if (nn == 0) *(v8f*)(pout + n0g + hi) = acc2;
    }
    __syncthreads();

    // ---- publish partial, device-scope barrier (double-buffered) ----
    const int buf = i & 1;
    float* mypart = gpart + ((size_t)buf * NW + g) * PDIM;
    for (int idx = tid; idx < PDIM; idx += 256) mypart[idx] = pout[idx];
    __threadfence();
    __syncthreads();
    if (tid == 0) {
      __hip_atomic_fetch_add(&ctr[buf], 1u, __ATOMIC_RELEASE, __HIP_MEMORY_SCOPE_AGENT);
      unsigned target = (unsigned)(i / 2 + 1) * NW;
      while (__hip_atomic_load(&ctr[buf], __ATOMIC_ACQUIRE, __HIP_MEMORY_SCOPE_AGENT) < target)
        __builtin_amdgcn_s_sleep(2);
    }
    __syncthreads();
    (void)__hip_atomic_load(&ctr[buf], __ATOMIC_ACQUIRE, __HIP_MEMORY_SCOPE_AGENT);

    // ---- cross-WGP reduce: rhs -> state update; WGP0 emits y ----
    const float* allp = gpart + (size_t)buf * NW * PDIM;
    float sr = 0.f;
#pragma unroll
    for (int g2 = 0; g2 < NW; ++g2) sr += allp[g2 * PDIM + tid];
    if (g == 0 && tid < YDIM) {
      float sy = 0.f;
#pragma unroll
      for (int g2 = 0; g2 < NW; ++g2) sy += allp[g2 * PDIM + SDIM + tid];
      yout[(size_t)i * YDIM + tid] = sy + b2ys[tid];
    }
    float xn = zs[tid] + DT_STEP * (sr + b2rs[tid]);
    zs[tid] = xn;
    __syncthreads();
  }
}

extern "C" void kernel_launch(void* const* d_in, const int* in_sizes, int n_in,
                              void* d_out, int out_size, void* d_ws, size_t ws_size,
                              hipStream_t stream) {
  const float* x_init = (const float*)d_in[0];
  const float* u      = (const float*)d_in[1];
  const float* W1     = (const float*)d_in[2];
  const float* b1     = (const float*)d_in[3];
  const float* W2r    = (const float*)d_in[4];
  const float* b2r    = (const float*)d_in[5];
  const float* W2y    = (const float*)d_in[6];
  const float* b2y    = (const float*)d_in[7];

  unsigned* ctr  = (unsigned*)d_ws;                       // 2 counters
  float* gpart   = (float*)((char*)d_ws + 256);           // [2][NW][384]

  init_sync<<<1, 32, 0, stream>>>(ctr);
  ode_scan<<<dim3(NW), dim3(256), 0, stream>>>(x_init, u, W1, b1, W2r, b2r, W2y, b2y,
                                               (float*)d_out, gpart, ctr);
}